// HyRes_8658654069050
// MI455X (gfx1250) — compile-verified
//
#include <hip/hip_runtime.h>
#include <hip/hip_bf16.h>
#include <math.h>

// ---------------- problem constants ----------------
#define H_  384
#define W_  384
#define K_  191
#define KP  192          // K padded to a multiple of 16 for WMMA tiling
#define N_  (H_ * W_)    // 147456
#define NTS 15           // SURE threshold grid size
#define LVLS 5

typedef float v2f __attribute__((ext_vector_type(2)));
typedef float v8f __attribute__((ext_vector_type(8)));

// db5 analysis filters (pywt convention); HI[t] = LO[9-t] * (-1)^t
__constant__ float c_lo[10] = {
    0.003335725285001549f, -0.012580751999015526f, -0.006241490213011705f,
    0.07757149384006515f,  -0.03224486958502952f,  -0.24229488706619015f,
    0.13842814590110342f,   0.7243085284385744f,    0.6038292697974729f,
    0.160102397974125f};
__constant__ float c_hi[10] = {
    0.160102397974125f,    -0.6038292697974729f,    0.7243085284385744f,
   -0.13842814590110342f,  -0.24229488706619015f,   0.03224486958502952f,
    0.07757149384006515f,   0.006241490213011705f, -0.012580751999015526f,
   -0.003335725285001549f};

__device__ __forceinline__ v8f wmma4(v2f a, v2f b, v8f c) {
    // D = A(16x4,f32) x B(4x16,f32) + C(16x16,f32)   [V_WMMA_F32_16X16X4_F32]
    return __builtin_amdgcn_wmma_f32_16x16x4_f32(
        /*neg_a=*/false, a, /*neg_b=*/false, b,
        /*c_mod=*/(short)0, c, /*reuse_a=*/false, /*reuse_b=*/false);
}

// ---- CDNA5 async global->LDS copy (ASYNCcnt-tracked, §4 async/tensor ISA) ---
__device__ __forceinline__ void lds_async_ld(float* lds_dst, const float* g_src) {
    // VDST = per-lane LDS byte address (low 32 bits of the shared aperture),
    // VADDR = per-lane 64-bit global address.
    unsigned l = (unsigned)(unsigned long long)lds_dst;
    unsigned long long g = (unsigned long long)g_src;
    asm volatile("global_load_async_to_lds_b32 %0, %1, off"
                 :: "v"(l), "v"(g) : "memory");
}
__device__ __forceinline__ void lds_async_wait() {
    asm volatile("s_wait_asynccnt 0" ::: "memory");
}

// ---------------- utility ----------------
__global__ void k_zero(float* p, long n) {
    long i = (long)blockIdx.x * blockDim.x + threadIdx.x;
    if (i < n) p[i] = 0.f;
}

// ---------------- G = X^T X, X: N x K_ row-major, G: KP x KP (f32 atomics) ---
// One wave32 per 16x16 output tile; split-K over N via grid.z + f32 atomics.
// Load indices are clamped (not masked): the contraction axis N is unpadded,
// so clamped garbage only lands in output row/col 191, dropped at the store.
__global__ void k_gram_wmma(const float* __restrict__ X, float* __restrict__ G,
                            int nslab) {
    const int jt = blockIdx.x, it = blockIdx.y, slab = blockIdx.z;
    const int lane = threadIdx.x;           // exactly one wave32
    const int m  = lane & 15;
    const int hi = lane >> 4;               // 0: K=0,1  1: K=2,3
    const int rowA = it * 16 + m;
    const int colB = jt * 16 + m;
    const int rowC = rowA < K_ ? rowA : K_ - 1;
    const int colC = colB < K_ ? colB : K_ - 1;
    v8f acc = {};
    const int per = N_ / nslab;             // nslab divides N_, 16 | per
    const int n0 = slab * per;
    const float* p = X + (long)(n0 + 2 * hi) * K_;
    for (int n = 0; n < per; n += 16) {
        __builtin_prefetch(p + 64L * K_, 0, 1);
#pragma unroll
        for (int u = 0; u < 4; ++u) {
            const float* q = p + (long)(4 * u) * K_;
            v2f a, b;
            a.x = q[rowC];      a.y = q[K_ + rowC];
            b.x = q[colC];      b.y = q[K_ + colC];
            acc = wmma4(a, b, acc);
        }
        p += 16L * K_;
    }
#pragma unroll
    for (int r = 0; r < 8; ++r) {
        const int row = it * 16 + r + hi * 8;
        const int col = jt * 16 + m;
        if (row < K_ && col < K_) atomicAdd(&G[row * KP + col], acc[r]);
    }
}

// ------- C(NxK_) = alpha * A(NxK_) @ B(KPxKP) + beta * A ; optional C^T out --
// Contraction axis is padded 191->192, but B's padded rows are zero, so the
// clamped (finite) A reads at k=191 contribute exactly 0: no masks, no branches.
__global__ void k_gemm_nk_kk(const float* __restrict__ A,
                             const float* __restrict__ B,
                             float* __restrict__ C,
                             float alpha, float beta, int transpose_out) {
    const int jt = blockIdx.x;              // 0..11 column tile
    const int m0 = blockIdx.y * 16;         // row block
    const int lane = threadIdx.x;
    const int m  = lane & 15;
    const int hi = lane >> 4;
    const int rowA = m0 + m;
    const float* Ar = A + (long)rowA * K_;
    const float* Bc = B + jt * 16 + m;
    v8f acc = {};
    for (int k = 0; k < KP; k += 16) {
#pragma unroll
        for (int u = 0; u < 4; ++u) {
            const int ka = k + 4 * u + 2 * hi;
            const int k0 = (ka     < K_) ? ka     : K_ - 1;
            const int k1 = (ka + 1 < K_) ? ka + 1 : K_ - 1;
            v2f a, b;
            a.x = Ar[k0];               a.y = Ar[k1];
            b.x = Bc[(long)ka * KP];    b.y = Bc[(long)(ka + 1) * KP];
            acc = wmma4(a, b, acc);
        }
    }
#pragma unroll
    for (int r = 0; r < 8; ++r) {
        const int row = m0 + r + hi * 8;
        const int col = jt * 16 + m;
        if (col < K_) {
            float v = alpha * acc[r];
            if (beta != 0.f) v += beta * A[(long)row * K_ + col];
            if (transpose_out) C[(long)col * N_ + row] = v;
            else               C[(long)row * K_ + col] = v;
        }
    }
}

// ---------------- Gauss-Jordan inverse of (G + 1e-6 I), 191x191 --------------
__global__ void k_invert(const float* __restrict__ G, float* __restrict__ A,
                         float* __restrict__ Inv) {
    const int t = threadIdx.x, NT = blockDim.x;
    __shared__ float rowA_s[K_], rowI_s[K_], colA_s[K_];
    __shared__ float piv;
    for (int idx = t; idx < K_ * K_; idx += NT) {
        int r = idx / K_, c = idx % K_;
        A[idx]   = G[r * KP + c] + (r == c ? 1e-6f : 0.f);
        Inv[idx] = (r == c) ? 1.f : 0.f;
    }
    __syncthreads();
    for (int p = 0; p < K_; ++p) {
        if (t == 0) piv = A[p * K_ + p];
        __syncthreads();
        const float ip = 1.f / piv;
        for (int c = t; c < K_; c += NT) {
            rowA_s[c] = A[p * K_ + c] * ip;
            rowI_s[c] = Inv[p * K_ + c] * ip;
        }
        for (int r = t; r < K_; r += NT) colA_s[r] = A[r * K_ + p];
        __syncthreads();
        for (int idx = t; idx < K_ * K_; idx += NT) {
            int r = idx / K_, c = idx % K_;
            if (r == p) { A[idx] = rowA_s[c]; Inv[idx] = rowI_s[c]; }
            else {
                A[idx]   -= colA_s[r] * rowA_s[c];
                Inv[idx] -= colA_s[r] * rowI_s[c];
            }
        }
        __syncthreads();
    }
}

// ------- betasT[k][i] = betas[i][k] (HySime per-band regression), padded -----
__global__ void k_betas(const float* __restrict__ rr /*KP stride*/,
                        const float* __restrict__ rri /*K_ stride*/,
                        float* __restrict__ betasT /*KP x KP, pre-zeroed*/) {
    const int i = blockIdx.x;               // band
    const float d = rri[i * K_ + i];
    for (int j = threadIdx.x; j < K_; j += blockDim.x) {
        float s = 0.f;
        const float rji = rri[j * K_ + i];
        for (int k = 0; k < K_; ++k) {
            float xx = rri[j * K_ + k] - rji * rri[i * K_ + k] / d;
            s += xx * rr[k * KP + i];
        }
        if (j == i) s = 0.f;
        betasT[j * KP + i] = s;
    }
}

// ---------------- per-band variance (ddof=1) of residual (NxK_) --------------
__global__ void k_colvar(const float* __restrict__ Wv, float* __restrict__ om) {
    const int b = blockIdx.x;
    __shared__ float ss[256], sq[256];
    float s = 0.f, q = 0.f;
    for (long n = threadIdx.x; n < N_; n += 256) {
        float v = Wv[n * K_ + b]; s += v; q += v * v;
    }
    ss[threadIdx.x] = s; sq[threadIdx.x] = q; __syncthreads();
    for (int st = 128; st > 0; st >>= 1) {
        if (threadIdx.x < st) { ss[threadIdx.x] += ss[threadIdx.x + st];
                                sq[threadIdx.x] += sq[threadIdx.x + st]; }
        __syncthreads();
    }
    if (threadIdx.x == 0) {
        float mean = ss[0] / (float)N_;
        om[b] = (sq[0] - (float)N_ * mean * mean) / (float)(N_ - 1);
    }
}

__global__ void k_whiten(const float* __restrict__ x,
                         const float* __restrict__ om, float* __restrict__ y) {
    long i = (long)blockIdx.x * 256 + threadIdx.x;
    if (i >= (long)N_ * K_) return;
    int b = (int)(i % K_);
    y[i] = x[i] * rsqrtf(om[b] + 1e-30f);
}

// ---------------- cyclic Jacobi eigensolver (SVD surrogate) ------------------
__global__ void k_jacobi(const float* __restrict__ G, float* __restrict__ A,
                         float* __restrict__ V, float* __restrict__ ev) {
    const int t = threadIdx.x, NT = blockDim.x;
    __shared__ float cs, sn;
    __shared__ int imax;
    for (int idx = t; idx < K_ * K_; idx += NT) {
        int r = idx / K_, c = idx % K_;
        A[idx] = G[r * KP + c];
        V[idx] = (r == c) ? 1.f : 0.f;
    }
    __syncthreads();
    for (int sweep = 0; sweep < 6; ++sweep)
        for (int p = 0; p < K_ - 1; ++p)
            for (int q = p + 1; q < K_; ++q) {
                if (t == 0) {
                    float apq = A[p * K_ + q];
                    if (fabsf(apq) < 1e-12f) { cs = 1.f; sn = 0.f; }
                    else {
                        float th = 0.5f * (A[q * K_ + q] - A[p * K_ + p]) / apq;
                        float tt = (th >= 0.f ? 1.f : -1.f) /
                                   (fabsf(th) + sqrtf(th * th + 1.f));
                        cs = rsqrtf(tt * tt + 1.f); sn = tt * cs;
                    }
                }
                __syncthreads();
                if (sn != 0.f) {
                    const float c0 = cs, s0 = sn;
                    for (int k = t; k < K_; k += NT) {
                        float akp = A[k * K_ + p], akq = A[k * K_ + q];
                        A[k * K_ + p] = c0 * akp - s0 * akq;
                        A[k * K_ + q] = s0 * akp + c0 * akq;
                    }
                    __syncthreads();
                    for (int k = t; k < K_; k += NT) {
                        float apk = A[p * K_ + k], aqk = A[q * K_ + k];
                        A[p * K_ + k] = c0 * apk - s0 * aqk;
                        A[q * K_ + k] = s0 * apk + c0 * aqk;
                        float vkp = V[k * K_ + p], vkq = V[k * K_ + q];
                        V[k * K_ + p] = c0 * vkp - s0 * vkq;
                        V[k * K_ + q] = s0 * vkp + c0 * vkq;
                    }
                }
                __syncthreads();
            }
    // sort eigenpairs descending (selection sort, column swaps)
    for (int i = 0; i < K_; ++i) {
        if (t == 0) {
            int mi = i; float mv = A[i * K_ + i];
            for (int j = i + 1; j < K_; ++j)
                if (A[j * K_ + j] > mv) { mv = A[j * K_ + j]; mi = j; }
            imax = mi;
        }
        __syncthreads();
        const int mi = imax;
        if (mi != i) {
            for (int k = t; k < K_; k += NT) {
                float tmp = V[k * K_ + i];
                V[k * K_ + i] = V[k * K_ + mi];
                V[k * K_ + mi] = tmp;
            }
            if (t == 0) {
                float tmp = A[i * K_ + i];
                A[i * K_ + i] = A[mi * K_ + mi];
                A[mi * K_ + mi] = tmp;
            }
        }
        __syncthreads();
    }
    for (int i = t; i < K_; i += NT) ev[i] = A[i * K_ + i];
}

// ------- pad 191x191 matrix into zeroed 192x192, optionally transposed -------
__global__ void k_pad(const float* __restrict__ V, float* __restrict__ B,
                      int transpose) {
    int idx = blockIdx.x * 256 + threadIdx.x;
    if (idx >= KP * KP) return;
    int r = idx / KP, c = idx % KP;
    float v = 0.f;
    if (r < K_ && c < K_) v = transpose ? V[c * K_ + r] : V[r * K_ + c];
    B[idx] = v;
}

// ------- periodized db5 DWT, per-channel, async-LDS staged -------------------
__global__ void k_dwt_cols(const float* __restrict__ in, float* __restrict__ out,
                           int S) {
    const int c = blockIdx.y, col = blockIdx.x;
    __shared__ float v[384];
    const float* src = in + (long)c * N_ + col;
    for (int i = threadIdx.x; i < S; i += blockDim.x)
        lds_async_ld(&v[i], &src[(long)i * W_]);
    lds_async_wait();
    __syncthreads();
    const int h = S / 2;
    float* dst = out + (long)c * N_ + col;
    for (int j = threadIdx.x; j < h; j += blockDim.x) {
        float lo = 0.f, hi = 0.f;
#pragma unroll
        for (int tt = 0; tt < 10; ++tt) {
            float x = v[(2 * j + tt) % S];
            lo += x * c_lo[tt]; hi += x * c_hi[tt];
        }
        dst[(long)j * W_] = lo; dst[(long)(h + j) * W_] = hi;
    }
}
__global__ void k_dwt_rows(const float* __restrict__ in, float* __restrict__ out,
                           int S) {
    const int c = blockIdx.y, r = blockIdx.x;
    __shared__ float v[384];
    const float* src = in + (long)c * N_ + (long)r * W_;
    for (int i = threadIdx.x; i < S; i += blockDim.x)
        lds_async_ld(&v[i], &src[i]);
    lds_async_wait();
    __syncthreads();
    const int h = S / 2;
    float* dst = out + (long)c * N_ + (long)r * W_;
    for (int j = threadIdx.x; j < h; j += blockDim.x) {
        float lo = 0.f, hi = 0.f;
#pragma unroll
        for (int tt = 0; tt < 10; ++tt) {
            float x = v[(2 * j + tt) % S];
            lo += x * c_lo[tt]; hi += x * c_hi[tt];
        }
        dst[j] = lo; dst[h + j] = hi;
    }
}
__global__ void k_idwt_rows(const float* __restrict__ in, float* __restrict__ out,
                            int S) {
    const int c = blockIdx.y, r = blockIdx.x;
    __shared__ float a[192], d[192];
    const int h = S / 2;
    const float* src = in + (long)c * N_ + (long)r * W_;
    for (int i = threadIdx.x; i < h; i += blockDim.x) {
        lds_async_ld(&a[i], &src[i]);
        lds_async_ld(&d[i], &src[h + i]);
    }
    lds_async_wait();
    __syncthreads();
    float* dst = out + (long)c * N_ + (long)r * W_;
    for (int m = threadIdx.x; m < S; m += blockDim.x) {
        float y = 0.f;
#pragma unroll
        for (int tt = 0; tt < 10; ++tt) {
            int idx = (m + tt - 9 + S) % S;
            if ((idx & 1) == 0) {
                int i2 = idx >> 1;
                y += a[i2] * c_lo[9 - tt] + d[i2] * c_hi[9 - tt];
            }
        }
        dst[m] = y;
    }
}
__global__ void k_idwt_cols(const float* __restrict__ in, float* __restrict__ out,
                            int S) {
    const int c = blockIdx.y, col = blockIdx.x;
    __shared__ float a[192], d[192];
    const int h = S / 2;
    const float* src = in + (long)c * N_ + col;
    for (int i = threadIdx.x; i < h; i += blockDim.x) {
        lds_async_ld(&a[i], &src[(long)i * W_]);
        lds_async_ld(&d[i], &src[(long)(h + i) * W_]);
    }
    lds_async_wait();
    __syncthreads();
    float* dst = out + (long)c * N_ + col;
    for (int m = threadIdx.x; m < S; m += blockDim.x) {
        float y = 0.f;
#pragma unroll
        for (int tt = 0; tt < 10; ++tt) {
            int idx = (m + tt - 9 + S) % S;
            if ((idx & 1) == 0) {
                int i2 = idx >> 1;
                y += a[i2] * c_lo[9 - tt] + d[i2] * c_hi[9 - tt];
            }
        }
        dst[(long)m * W_] = y;
    }
}

// ---------------- SURE rank selection ----------------------------------------
__global__ void k_colnorm(const float* __restrict__ img, float* __restrict__ cn) {
    const int b = blockIdx.x;
    __shared__ float sm[256];
    float s = 0.f;
    for (long n = threadIdx.x; n < N_; n += 256) {
        float v = img[(long)b * N_ + n]; s += v * v;
    }
    sm[threadIdx.x] = s; __syncthreads();
    for (int st = 128; st > 0; st >>= 1) {
        if (threadIdx.x < st) sm[threadIdx.x] += sm[threadIdx.x + st];
        __syncthreads();
    }
    if (threadIdx.x == 0) cn[b] = sm[0];
}
__global__ void k_sure(const float* __restrict__ img,
                       const float* __restrict__ cn, float* __restrict__ sure) {
    const int b = blockIdx.x, ti = blockIdx.y;
    const float tmax = sqrtf(logf((float)N_));
    const float t = tmax * (float)ti / (float)(NTS - 1);
    const float t2 = t * t;
    __shared__ float sm[256];
    float acc = 0.f;
    for (long n = threadIdx.x; n < N_; n += 256) {
        float v = img[(long)b * N_ + n]; float v2 = v * v;
        acc += fminf(v2, t2) + 2.f * (v2 > t2 ? 1.f : 0.f);
    }
    sm[threadIdx.x] = acc; __syncthreads();
    for (int st = 128; st > 0; st >>= 1) {
        if (threadIdx.x < st) sm[threadIdx.x] += sm[threadIdx.x + st];
        __syncthreads();
    }
    if (threadIdx.x == 0) sure[b * NTS + ti] = sm[0] - cn[b];
}
__global__ void k_rank(const float* __restrict__ cn,
                       const float* __restrict__ sure, int* __restrict__ rank) {
    if (threadIdx.x != 0) return;
    float normv = 0.f;
    for (int b = 0; b < K_; ++b) normv += cn[b];
    const float cconst = normv - (float)((long)N_ * K_);
    float acc[NTS]; float minsure[K_];
    for (int i = 0; i < NTS; ++i) acc[i] = 0.f;
    for (int r = 0; r < K_; ++r) {
        float mn = 3.4e38f;
        for (int i = 0; i < NTS; ++i) {
            float s = acc[i] + sure[r * NTS + i] + cconst;
            acc[i] += s;
            if (s < mn) mn = s;
        }
        minsure[r] = mn;
    }
    int rk = K_ - 1;
    for (int r = 2; r < K_; ++r)
        if (minsure[r] > minsure[r - 1]) { rk = r; break; }
    *rank = rk;
}
__global__ void k_mask(float* __restrict__ img, const int* __restrict__ rank) {
    long i = (long)blockIdx.x * 256 + threadIdx.x;
    if (i >= (long)N_ * K_) return;
    int ch = (int)(i / N_);
    if (ch >= *rank) img[i] = 0.f;
}

// ---------------- layout shuffles + final scaling ----------------------------
__global__ void k_transpose_kn_nk(const float* __restrict__ img,
                                  float* __restrict__ out) {
    long i = (long)blockIdx.x * 256 + threadIdx.x;
    if (i >= (long)N_ * K_) return;
    long n = i / K_; int b = (int)(i % K_);
    out[i] = img[(long)b * N_ + n];
}
__global__ void k_scale_out(float* __restrict__ out,
                            const float* __restrict__ om) {
    long i = (long)blockIdx.x * 256 + threadIdx.x;
    if (i >= (long)N_ * K_) return;
    int b = (int)(i % K_);
    out[i] *= sqrtf(om[b]);
}

// =============================================================================
extern "C" void kernel_launch(void* const* d_in, const int* in_sizes, int n_in,
                              void* d_out, int out_size, void* d_ws, size_t ws_size,
                              hipStream_t stream) {
    const float* x = (const float*)d_in[0];   // (H,W,K) f32 == (N_,K_) row-major
    float* out = (float*)d_out;

    // workspace carve-out
    float* ws  = (float*)d_ws;
    const long NK = (long)N_ * K_;
    float* yw   = ws;                 // N_ x K_   whitened data
    float* t1   = yw  + NK;           // N_ x K_   GEMM temp / residual / y_est
    float* img  = t1  + NK;           // K_ x N_   channel-major coeff image
    float* img2 = img + NK;           // K_ x N_   DWT ping-pong
    float* M0   = img2 + NK;          // KP x KP   Gram
    float* M1   = M0 + KP * KP;       // K_ x K_   inverse
    float* M2   = M1 + KP * KP;       // KP x KP   betas^T (padded)
    float* M3   = M2 + KP * KP;       // K_ x K_   eigenvectors V
    float* M4   = M3 + KP * KP;       // KP x KP   padded V / V^T
    float* M5   = M4 + KP * KP;       // K_ x K_   GJ / Jacobi scratch
    float* evals   = M5 + KP * KP;    // KP
    float* omega   = evals + KP;      // KP
    float* colnorm = omega + KP;      // KP
    float* sure    = colnorm + KP;    // K_ x 15
    int*   rank    = (int*)(sure + KP * 16);

    const int SLABS = 144;            // 1024 rows of N per slab (16 | 1024)
    const dim3 gG(12, 12, SLABS);     // Gram tiles x split-K
    const dim3 gM(12, N_ / 16);       // NxK GEMM tiles
    const int BZ = (int)((NK + 255) / 256);

    // 1) rr = X^T X  (f32 WMMA, split-K atomics)
    k_zero<<<(KP * KP + 255) / 256, 256, 0, stream>>>(M0, KP * KP);
    k_gram_wmma<<<gG, 32, 0, stream>>>(x, M0, SLABS);
    // 2) rri = (rr + 1e-6 I)^-1 ; betas
    k_invert<<<1, 256, 0, stream>>>(M0, M5, M1);
    k_zero<<<(KP * KP + 255) / 256, 256, 0, stream>>>(M2, KP * KP);
    k_betas<<<K_, 256, 0, stream>>>(M0, M1, M2);
    // 3) W = X - X betas^T  (WMMA, alpha=-1 beta=1) -> variance -> whiten
    k_gemm_nk_kk<<<gM, 32, 0, stream>>>(x, M2, t1, -1.f, 1.f, 0);
    k_colvar<<<K_, 256, 0, stream>>>(t1, omega);
    k_whiten<<<BZ, 256, 0, stream>>>(x, omega, yw);
    // 4) G2 = Y^T Y -> Jacobi eig (SVD surrogate) -> pc = Y V  (channel-major)
    k_zero<<<(KP * KP + 255) / 256, 256, 0, stream>>>(M0, KP * KP);
    k_gram_wmma<<<gG, 32, 0, stream>>>(yw, M0, SLABS);
    k_jacobi<<<1, 256, 0, stream>>>(M0, M5, M3, evals);
    k_pad<<<(KP * KP + 255) / 256, 256, 0, stream>>>(M3, M4, 0);
    k_gemm_nk_kk<<<gM, 32, 0, stream>>>(yw, M4, img, 1.f, 0.f, /*transpose*/1);
    // 5) 5-level db5 2D DWT per channel (cols then rows, in place on region)
    int S = 384;
    for (int l = 0; l < LVLS; ++l) {
        k_dwt_cols<<<dim3(S, K_), 128, 0, stream>>>(img, img2, S);
        k_dwt_rows<<<dim3(S, K_), 128, 0, stream>>>(img2, img, S);
        S >>= 1;
    }
    // 6) SURE rank selection + channel mask
    k_colnorm<<<K_, 256, 0, stream>>>(img, colnorm);
    k_sure<<<dim3(K_, NTS), 256, 0, stream>>>(img, colnorm, sure);
    k_rank<<<1, 32, 0, stream>>>(colnorm, sure, rank);
    k_mask<<<BZ, 256, 0, stream>>>(img, rank);
    // 7) inverse DWT (rows then cols, small -> large)
    for (int s2 = 24; s2 <= 384; s2 <<= 1) {
        k_idwt_rows<<<dim3(s2, K_), 128, 0, stream>>>(img, img2, s2);
        k_idwt_cols<<<dim3(s2, K_), 128, 0, stream>>>(img2, img, s2);
    }
    // 8) out = sqrt(omega) * (y_est @ V^T)
    k_transpose_kn_nk<<<BZ, 256, 0, stream>>>(img, t1);
    k_pad<<<(KP * KP + 255) / 256, 256, 0, stream>>>(M3, M4, 1);
    k_gemm_nk_kk<<<gM, 32, 0, stream>>>(t1, M4, out, 1.f, 0.f, 0);
    k_scale_out<<<BZ, 256, 0, stream>>>(out, omega);
    (void)in_sizes; (void)n_in; (void)out_size; (void)ws_size;
}